// TorchHawkes_26293789786925
// MI455X (gfx1250) — compile-verified
//
#include <hip/hip_runtime.h>
#include <hip/hip_bf16.h>
#include <math.h>

// Problem constants (reference: K=1024, N=16384)
#define KDIM 1024
#define NDIM 16384
#define CHUNK 128           // scan chunk length
#define NCHUNK (NDIM / CHUNK)

// GEMM tiling
#define BM 64
#define BN 128
#define BK 16
#define LDSPAD 20           // row stride in floats (TDM pads 4 dwords every 16)

typedef __attribute__((ext_vector_type(2))) float v2f;
typedef __attribute__((ext_vector_type(8))) float v8f;
typedef __attribute__((ext_vector_type(4))) unsigned int v4u;
typedef __attribute__((ext_vector_type(8))) int v8i_t;
typedef __attribute__((ext_vector_type(4))) int v4i_t;

// ---------------------------------------------------------------------------
// hmu[k] = mean(obs[k,:]) / 10 + 0.01
__global__ __launch_bounds__(256) void hmu_kernel(const float* __restrict__ obs,
                                                  float* __restrict__ hmu) {
    __shared__ float red[256];
    const int k = blockIdx.x;
    const float4* row = (const float4*)&obs[(size_t)k * NDIM];
    float s = 0.f;
    for (int j = threadIdx.x; j < NDIM / 4; j += 256) {
        float4 v = row[j];
        s += (v.x + v.y) + (v.z + v.w);
    }
    red[threadIdx.x] = s;
    __syncthreads();
    for (int st = 128; st > 0; st >>= 1) {
        if (threadIdx.x < st) red[threadIdx.x] += red[threadIdx.x + st];
        __syncthreads();
    }
    if (threadIdx.x == 0)
        hmu[k] = red[0] * (1.0f / NDIM) * 0.1f + 0.01f;
}

// ---------------------------------------------------------------------------
// Phase A: per-chunk carry with zero initial state.  S <- decay*(S + beta*obs_t)
__global__ __launch_bounds__(256) void carry_kernel(const float* __restrict__ obs,
                                                    const float* __restrict__ hbeta,
                                                    float* __restrict__ carry) {
    const int c = blockIdx.x;
    const int k = blockIdx.y * 256 + threadIdx.x;
    const float beta = hbeta[0];
    const float decay = expf(-beta);
    const float4* p = (const float4*)&obs[(size_t)k * NDIM + c * CHUNK];
    float S = 0.f;
#pragma unroll 4
    for (int i = 0; i < CHUNK / 4; ++i) {
        float4 v = p[i];
        S = decay * (S + beta * v.x);
        S = decay * (S + beta * v.y);
        S = decay * (S + beta * v.z);
        S = decay * (S + beta * v.w);
    }
    carry[c * KDIM + k] = S;
}

// Phase B: serial affine composition across chunks.
__global__ __launch_bounds__(256) void prefix_kernel(const float* __restrict__ hbeta,
                                                     const float* __restrict__ carry,
                                                     float* __restrict__ prefix) {
    const int k = blockIdx.x * 256 + threadIdx.x;
    const float beta = hbeta[0];
    const float dL = expf(-beta * (float)CHUNK);   // decay^CHUNK
    float P = 0.f;
    for (int c = 0; c < NCHUNK; ++c) {
        prefix[c * KDIM + k] = P;
        P = dL * P + carry[c * KDIM + k];
    }
}

// Phase C: rescan with chunk-entry state; S_all[t,k] = S *before* obs_t.
__global__ __launch_bounds__(256) void scan_kernel(const float* __restrict__ obs,
                                                   const float* __restrict__ hbeta,
                                                   const float* __restrict__ prefix,
                                                   float* __restrict__ S_all) {
    const int c = blockIdx.x;
    const int k = blockIdx.y * 256 + threadIdx.x;
    const float beta = hbeta[0];
    const float decay = expf(-beta);
    float S = prefix[c * KDIM + k];
    const float4* p = (const float4*)&obs[(size_t)k * NDIM + c * CHUNK];
    const int tb = c * CHUNK;
#pragma unroll 4
    for (int i = 0; i < CHUNK / 4; ++i) {
        float4 v = p[i];
        S_all[(size_t)(tb + 4 * i + 0) * KDIM + k] = S; S = decay * (S + beta * v.x);
        S_all[(size_t)(tb + 4 * i + 1) * KDIM + k] = S; S = decay * (S + beta * v.y);
        S_all[(size_t)(tb + 4 * i + 2) * KDIM + k] = S; S = decay * (S + beta * v.z);
        S_all[(size_t)(tb + 4 * i + 3) * KDIM + k] = S; S = decay * (S + beta * v.w);
    }
}

// ---------------------------------------------------------------------------
// TDM: load a (rows x 16 f32) tile with row stride `stride_elems` into LDS at
// byte offset `lds_off`, padding LDS rows 16->20 dwords (conflict-free layout).
// D# layout per CDNA5 ISA ch.8 (groups 0/1; groups 2/3 zero for 2-D tensors).
// This toolchain exposes the 6-arg builtin form:
//   (uint32x4 g0, int32x8 g1, int32x4, int32x4, int32x8, i32 cpol)
__device__ __forceinline__ void tdm_load_tile(unsigned int lds_off, const float* gsrc,
                                              int rows, int stride_elems) {
    const unsigned long long ga = (unsigned long long)(uintptr_t)gsrc;
    v4u g0;
    g0[0] = 1u;                                               // count=1 (valid), user mode
    g0[1] = lds_off;                                          // lds_addr (bytes)
    g0[2] = (unsigned int)ga;                                 // global_addr[31:0]
    g0[3] = (unsigned int)((ga >> 32) & 0x1FFFFFFu) | (2u << 30); // addr[56:32], type=2
    v8i_t g1;
    g1[0] = (2 << 16)        // data_size = 4B
          | (1 << 20)        // pad_enable
          | (3 << 22)        // pad_interval: 16 dwords
          | (3 << 25);       // pad_amount: 4 dwords
    g1[1] = (16 << 16);      // tensor_dim0 = 16 (tile is interior)
    g1[2] = (rows << 16);    // tensor_dim1[15:0]
    g1[3] = (16 << 16);      // tile_dim0 = 16
    g1[4] = rows;            // tile_dim1
    g1[5] = stride_elems;    // tensor_dim0_stride[31:0]
    g1[6] = 0;
    g1[7] = 0;
    v4i_t z4 = {0, 0, 0, 0};
    v8i_t z8 = {0, 0, 0, 0, 0, 0, 0, 0};
    __builtin_amdgcn_tensor_load_to_lds(g0, g1, z4, z4, z8, 0);
}

// ---------------------------------------------------------------------------
// Fused GEMM (lam1 = halpha @ S_all^T) + softplus + lams store + partial loglik.
// 256 thr (8 wave32); BM x BN x BK LDS tiles, double-buffered, staged by the TDM
// (wave 0 issues tensor_load_to_lds; completion via s_wait_tensorcnt + barrier).
__global__ __launch_bounds__(256) void gemm_kernel(const float* __restrict__ halpha,
                                                   const float* __restrict__ S_all,
                                                   const float* __restrict__ hmu,
                                                   const float* __restrict__ obs,
                                                   float* __restrict__ out,
                                                   float* __restrict__ partials) {
    __shared__ __align__(16) float As[2][BM][LDSPAD];
    __shared__ __align__(16) float Bs[2][BN][LDSPAD];
    __shared__ float red[256];

    const int tid = threadIdx.x;
    const int n0 = blockIdx.x * BN;
    const int m0 = blockIdx.y * BM;
    const int lane = tid & 31;
    const int w    = tid >> 5;       // 0..7
    const int wm   = w & 1;          // M wave index (0..1) -> 32 rows each
    const int wn   = w >> 1;         // N wave index (0..3) -> 32 cols each
    const int l15  = lane & 15;
    const int hi   = lane >> 4;      // half-wave select

    // LDS byte offsets of the two buffers (flat low-32 == LDS offset).
    const unsigned int asOff0 = (unsigned int)(uintptr_t)&As[0][0][0];
    const unsigned int asOff1 = (unsigned int)(uintptr_t)&As[1][0][0];
    const unsigned int bsOff0 = (unsigned int)(uintptr_t)&Bs[0][0][0];
    const unsigned int bsOff1 = (unsigned int)(uintptr_t)&Bs[1][0][0];

    v8f acc[2][2];
#pragma unroll
    for (int mi = 0; mi < 2; ++mi)
#pragma unroll
        for (int ni = 0; ni < 2; ++ni) acc[mi][ni] = {};

    // Preload buffer 0 (TDM, wave 0 only; EXEC uniform within the wave).
    if (w == 0) {
        tdm_load_tile(asOff0, &halpha[(size_t)m0 * KDIM], BM, KDIM);
        tdm_load_tile(bsOff0, &S_all[(size_t)n0 * KDIM], BN, KDIM);
        __builtin_amdgcn_s_wait_tensorcnt(0);
    }
    __syncthreads();

    int cur = 0;
    for (int it = 0; it < KDIM / BK; ++it) {
        const int kb = it * BK;
        // Kick off next tiles into the other buffer while we compute.
        if (kb + BK < KDIM && w == 0) {
            tdm_load_tile(cur ? asOff0 : asOff1, &halpha[(size_t)m0 * KDIM + kb + BK], BM, KDIM);
            tdm_load_tile(cur ? bsOff0 : bsOff1, &S_all[(size_t)n0 * KDIM + kb + BK], BN, KDIM);
        }

#pragma unroll
        for (int ks = 0; ks < BK / 4; ++ks) {
            // ISA layouts: A lane -> (row=l15, k=ks*4+2*hi, +1);
            //              B lane -> (k=ks*4+2*hi, +1, col=l15)
            v2f a0 = *(const v2f*)&As[cur][wm * 32 +      l15][ks * 4 + hi * 2];
            v2f a1 = *(const v2f*)&As[cur][wm * 32 + 16 + l15][ks * 4 + hi * 2];
            v2f b0 = *(const v2f*)&Bs[cur][wn * 32 +      l15][ks * 4 + hi * 2];
            v2f b1 = *(const v2f*)&Bs[cur][wn * 32 + 16 + l15][ks * 4 + hi * 2];
            acc[0][0] = __builtin_amdgcn_wmma_f32_16x16x4_f32(false, a0, false, b0, (short)0, acc[0][0], false, false);
            acc[0][1] = __builtin_amdgcn_wmma_f32_16x16x4_f32(false, a0, false, b1, (short)0, acc[0][1], false, false);
            acc[1][0] = __builtin_amdgcn_wmma_f32_16x16x4_f32(false, a1, false, b0, (short)0, acc[1][0], false, false);
            acc[1][1] = __builtin_amdgcn_wmma_f32_16x16x4_f32(false, a1, false, b1, (short)0, acc[1][1], false, false);
        }

        // Issuing wave waits for the TDM, then everyone syncs before swapping.
        if (w == 0) __builtin_amdgcn_s_wait_tensorcnt(0);
        __syncthreads();
        cur ^= 1;
    }

    // Epilogue: lams = softplus(hmu + lam1); accumulate obs*log(lams) - lams.
    float lsum = 0.f;
#pragma unroll
    for (int mi = 0; mi < 2; ++mi) {
#pragma unroll
        for (int ni = 0; ni < 2; ++ni) {
            const int gn = n0 + wn * 32 + ni * 16 + l15;
#pragma unroll
            for (int r = 0; r < 8; ++r) {
                const int gm = m0 + wm * 32 + mi * 16 + r + 8 * hi;   // C layout: row r+8*hi
                const float x  = acc[mi][ni][r] + hmu[gm];
                const float sp = fmaxf(x, 0.f) + log1pf(expf(-fabsf(x)));  // stable softplus
                out[1 + (size_t)gm * NDIM + gn] = sp;
                const float o = obs[(size_t)gm * NDIM + gn];
                lsum += o * logf(sp) - sp;
            }
        }
    }
    red[tid] = lsum;
    __syncthreads();
    for (int st = 128; st > 0; st >>= 1) {
        if (tid < st) red[tid] += red[tid + st];
        __syncthreads();
    }
    if (tid == 0) partials[blockIdx.y * gridDim.x + blockIdx.x] = red[0];
}

// Deterministic final reduction of per-block partials -> out[0].
__global__ __launch_bounds__(256) void reduce_kernel(const float* __restrict__ partials,
                                                     int n, float* __restrict__ out) {
    __shared__ float red[256];
    float s = 0.f;
    for (int i = threadIdx.x; i < n; i += 256) s += partials[i];
    red[threadIdx.x] = s;
    __syncthreads();
    for (int st = 128; st > 0; st >>= 1) {
        if (threadIdx.x < st) red[threadIdx.x] += red[threadIdx.x + st];
        __syncthreads();
    }
    if (threadIdx.x == 0) out[0] = red[0];
}

// ---------------------------------------------------------------------------
extern "C" void kernel_launch(void* const* d_in, const int* in_sizes, int n_in,
                              void* d_out, int out_size, void* d_ws, size_t ws_size,
                              hipStream_t stream) {
    const float* obs    = (const float*)d_in[0];   // [K, N]
    const float* halpha = (const float*)d_in[1];   // [K, K]
    const float* hbeta  = (const float*)d_in[2];   // [1]
    float* out = (float*)d_out;                    // [1 + K*N]: loglik, lams

    float* ws       = (float*)d_ws;
    float* S_all    = ws;                                   // N*K floats (64 MB)
    float* hmu      = S_all + (size_t)NDIM * KDIM;          // K
    float* carry    = hmu + KDIM;                           // NCHUNK*K
    float* prefix   = carry + NCHUNK * KDIM;                // NCHUNK*K
    float* partials = prefix + NCHUNK * KDIM;               // grid-size

    hmu_kernel<<<KDIM, 256, 0, stream>>>(obs, hmu);
    carry_kernel<<<dim3(NCHUNK, KDIM / 256), 256, 0, stream>>>(obs, hbeta, carry);
    prefix_kernel<<<KDIM / 256, 256, 0, stream>>>(hbeta, carry, prefix);
    scan_kernel<<<dim3(NCHUNK, KDIM / 256), 256, 0, stream>>>(obs, hbeta, prefix, S_all);

    const dim3 ggrid(NDIM / BN, KDIM / BM);  // 128 x 16 = 2048 blocks
    gemm_kernel<<<ggrid, 256, 0, stream>>>(halpha, S_all, hmu, obs, out, partials);
    reduce_kernel<<<1, 256, 0, stream>>>(partials, (NDIM / BN) * (KDIM / BM), out);
}